// NonLocalAggregation_38989713113484
// MI455X (gfx1250) — compile-verified
//
#include <hip/hip_runtime.h>
#include <hip/hip_bf16.h>

typedef __attribute__((ext_vector_type(2))) float v2f;
typedef __attribute__((ext_vector_type(8))) float v8f;

#define NPIX 1024
#define NCH  32
#define TM   32      // rows per workgroup tile
#define PX   33      // xAll / weight / diff pitch (bank-conflict-free transpose)
#define PD   1026    // D-tile pitch: 1026 % 8 == 2 -> disjoint half-wave store banks
#define NEG_BIG (-3.4e38f)

__global__ __launch_bounds__(256, 1)
void nonlocal_agg_kernel(const float* __restrict__ x,
                         const float* __restrict__ lm,
                         const float* __restrict__ Wd,
                         const float* __restrict__ bd,
                         const float* __restrict__ Ws,
                         const float* __restrict__ bs,
                         const float* __restrict__ bias,
                         float* __restrict__ out)
{
    __shared__ float xAll[NPIX * PX];   // whole batch xf, row-major (n, c), 132 KB
    __shared__ float sq[NPIX];          // squared norms r
    __shared__ float Dt[TM * PD];       // 32x1024 distance tile; reused as OutBuf
    __shared__ float DiffBuf[TM * PX];  // xf - mean_sel for this tile
    __shared__ float WdL[NCH * PX];
    __shared__ float WsL[NCH * PX];
    __shared__ float bbL[NCH];

    const int t    = threadIdx.x;
    const int lane = t & 31;
    const int wave = t >> 5;
    const int b    = blockIdx.x >> 5;     // 32 row-tiles per batch
    const int tile = blockIdx.x & 31;
    const int row0 = tile * TM;
    const float* xb = x + b * (NCH * NPIX);

    // ---- Stage 0: fill LDS (transpose x[b] from (c,n) to xAll[n][c]) ----
    for (int i = t; i < NCH * NPIX; i += 256) {
        int c = i >> 10;        // channel
        int n = i & 1023;       // pixel
        xAll[n * PX + c] = xb[i];
    }
    for (int i = t; i < NCH * NCH; i += 256) {
        int o = i >> 5, c = i & 31;
        WdL[o * PX + c] = Wd[i];
        WsL[o * PX + c] = Ws[i];
    }
    if (t < NCH) bbL[t] = bd[t] + bs[t] + bias[t];
    __syncthreads();

    // ---- Stage 1: squared norms ----
    for (int j = 0; j < 4; ++j) {
        int n = t + j * 256;
        float s = 0.f;
        #pragma unroll
        for (int c = 0; c < NCH; ++c) { float v = xAll[n * PX + c]; s += v * v; }
        sq[n] = s;
    }
    __syncthreads();

    // ---- Stage 2: distance tile via fp32 WMMA:  D = 2*X*X^T - r_i - r_j, masked ----
    {
        const int r16 = lane & 15;
        const int hi  = lane >> 4;          // which K-pair half this lane holds
        // Preload A fragments (both 16-row halves of the tile, all 8 K-chunks)
        v2f afr[2][8];
        #pragma unroll
        for (int m = 0; m < 2; ++m) {
            #pragma unroll
            for (int kk = 0; kk < 8; ++kk) {
                int row = row0 + m * 16 + r16;
                int kb  = 4 * kk + 2 * hi;
                afr[m][kk].x = xAll[row * PX + kb];
                afr[m][kk].y = xAll[row * PX + kb + 1];
            }
        }
        for (int jt = 0; jt < 8; ++jt) {
            int j = wave + jt * 8;          // column tile 0..63 (16 cols each)
            int colbase = j * 16;
            v2f bfr[8];
            #pragma unroll
            for (int kk = 0; kk < 8; ++kk) {
                int col = colbase + r16;
                int kb  = 4 * kk + 2 * hi;
                bfr[kk].x = xAll[col * PX + kb];
                bfr[kk].y = xAll[col * PX + kb + 1];
            }
            #pragma unroll
            for (int m = 0; m < 2; ++m) {
                v8f c = {};
                #pragma unroll
                for (int kk = 0; kk < 8; ++kk) {
                    c = __builtin_amdgcn_wmma_f32_16x16x4_f32(
                            false, afr[m][kk], false, bfr[kk],
                            (short)0, c, false, false);
                }
                int colG = colbase + r16;
                float rB = sq[colG];
                #pragma unroll
                for (int v = 0; v < 8; ++v) {
                    int M    = v + 8 * hi;
                    int rloc = m * 16 + M;
                    int rowG = row0 + rloc;
                    float d  = 2.f * c[v] - sq[rowG] - rB;
                    float mk = lm[rowG * NPIX + colG];
                    d = d * mk - (1.f - mk);
                    Dt[rloc * PD + colG] = d;
                }
            }
        }
    }
    __syncthreads();

    // ---- Stage 3: per-row top-8 (register-resident iterative argmax) + mean/diff ----
    for (int rr = 0; rr < 4; ++rr) {
        int rloc = wave * 4 + rr;
        int rowG = row0 + rloc;
        float val[32];
        #pragma unroll
        for (int jj = 0; jj < 32; ++jj) val[jj] = Dt[rloc * PD + lane + jj * 32];
        int found[8];
        #pragma unroll
        for (int it = 0; it < 8; ++it) {
            float bv = NEG_BIG; int bi = 0x7fffffff;
            #pragma unroll
            for (int jj = 0; jj < 32; ++jj) {
                int idx = lane + jj * 32;
                bool bt = (val[jj] > bv) || (val[jj] == bv && idx < bi);
                bv = bt ? val[jj] : bv;
                bi = bt ? idx : bi;
            }
            #pragma unroll
            for (int off = 16; off > 0; off >>= 1) {
                float ov = __shfl_xor(bv, off, 32);
                int   oi = __shfl_xor(bi, off, 32);
                bool bt = (ov > bv) || (ov == bv && oi < bi);
                bv = bt ? ov : bv;
                bi = bt ? oi : bi;
            }
            found[it] = bi;                 // broadcast to all lanes
            #pragma unroll
            for (int jj = 0; jj < 32; ++jj)
                if (lane + jj * 32 == bi) val[jj] = NEG_BIG;
        }
        // lane == channel: mean of 8 neighbors, then diff = xf - mean
        float msum = 0.f;
        #pragma unroll
        for (int it = 0; it < 8; ++it) msum += xAll[found[it] * PX + lane];
        DiffBuf[rloc * PX + lane] = xAll[rowG * PX + lane] - msum * 0.125f;
    }
    __syncthreads();

    // ---- Stage 4: out = Diff @ Wd^T + X @ Ws^T   (WMMA, 4 waves, one 16x16 tile each)
    if (wave < 4) {
        const int r16 = lane & 15;
        const int hi  = lane >> 4;
        const int mtile = wave >> 1, ntile = wave & 1;
        v8f c = {};
        #pragma unroll
        for (int kk = 0; kk < 8; ++kk) {
            int kb = 4 * kk + 2 * hi;
            v2f a, w;
            a.x = DiffBuf[(mtile * 16 + r16) * PX + kb];
            a.y = DiffBuf[(mtile * 16 + r16) * PX + kb + 1];
            w.x = WdL[(ntile * 16 + r16) * PX + kb];
            w.y = WdL[(ntile * 16 + r16) * PX + kb + 1];
            c = __builtin_amdgcn_wmma_f32_16x16x4_f32(false, a, false, w,
                                                      (short)0, c, false, false);
        }
        #pragma unroll
        for (int kk = 0; kk < 8; ++kk) {
            int kb = 4 * kk + 2 * hi;
            v2f a, w;
            a.x = xAll[(row0 + mtile * 16 + r16) * PX + kb];
            a.y = xAll[(row0 + mtile * 16 + r16) * PX + kb + 1];
            w.x = WsL[(ntile * 16 + r16) * PX + kb];
            w.y = WsL[(ntile * 16 + r16) * PX + kb + 1];
            c = __builtin_amdgcn_wmma_f32_16x16x4_f32(false, a, false, w,
                                                      (short)0, c, false, false);
        }
        #pragma unroll
        for (int v = 0; v < 8; ++v) {
            int M = v + 8 * hi;
            Dt[(mtile * 16 + M) * PX + ntile * 16 + r16] = c[v];  // Dt reused as OutBuf
        }
    }
    __syncthreads();

    // ---- Stage 5: coalesced NCHW store ----
    for (int j = 0; j < 4; ++j) {
        int i = t + j * 256;
        int o = i >> 5, r = i & 31;
        out[b * (NCH * NPIX) + o * NPIX + row0 + r] = Dt[r * PX + o] + bbL[o];
    }
}

extern "C" void kernel_launch(void* const* d_in, const int* in_sizes, int n_in,
                              void* d_out, int out_size, void* d_ws, size_t ws_size,
                              hipStream_t stream) {
    const float* x    = (const float*)d_in[0];
    const float* lm   = (const float*)d_in[1];
    const float* Wd   = (const float*)d_in[2];
    const float* bd   = (const float*)d_in[3];
    const float* Ws   = (const float*)d_in[4];
    const float* bs   = (const float*)d_in[5];
    const float* bias = (const float*)d_in[6];
    // d_in[7] is k == 8 (hardcoded in the kernel's top-8 extraction)
    float* out = (float*)d_out;
    (void)in_sizes; (void)n_in; (void)out_size; (void)d_ws; (void)ws_size;
    nonlocal_agg_kernel<<<dim3(32 * 32), dim3(256), 0, stream>>>(
        x, lm, Wd, bd, Ws, bs, bias, out);
}